// CoarseToFine_48060684043075
// MI455X (gfx1250) — compile-verified
//
#include <hip/hip_runtime.h>
#include <hip/hip_bf16.h>
#include <cstdint>
#include <cstddef>

typedef __attribute__((ext_vector_type(2))) float v2f;
typedef __attribute__((ext_vector_type(8))) float v8f;

#define WIN       5
#define WW        25       // WIN*WIN
#define STRIDE_U  4
#define PAD_U     2
#define CFINE     128
#define CCOARSE   256
#define HF        240
#define WF        320
#define WOUT      80
#define NM        2048     // M (number of matches)
#define LDS_STRIDE 68      // 64 k-cols + 4 pad -> bank-conflict-free, 16B-aligned rows

// ---------------------------------------------------------------------------
// Kernel A: c_proj[4096,128] = gather(feat_c0 / feat_c1) @ W_proj^T + b_proj
// ---------------------------------------------------------------------------
__global__ void __launch_bounds__(256)
proj_gemm_kernel(const float* __restrict__ feat_c0,
                 const float* __restrict__ feat_c1,
                 const int*   __restrict__ b_ids,
                 const int*   __restrict__ l_ids,
                 const int*   __restrict__ s_ids,
                 const float* __restrict__ Wp,
                 const float* __restrict__ bp,
                 float*       __restrict__ cproj,
                 int L, int S)
{
    __shared__ float wlds[128 * LDS_STRIDE];

    const int tid   = threadIdx.x;
    const int wv    = tid >> 5;
    const int lane  = tid & 31;
    const int l16   = lane & 15;
    const int hi    = lane >> 4;           // half-wave selector
    const int rbase = blockIdx.x * 128 + wv * 16;
    const int row   = rbase + l16;         // [0, 4096)

    // Gathered A row (256 contiguous floats)
    const float* arow;
    if (row < NM) {
        arow = feat_c0 + ((size_t)b_ids[row] * (size_t)L + (size_t)l_ids[row]) * CCOARSE;
    } else {
        const int mm = row - NM;
        arow = feat_c1 + ((size_t)b_ids[mm] * (size_t)S + (size_t)s_ids[mm]) * CCOARSE;
    }

    v8f acc[8];
    const v8f zero = {};
#pragma unroll
    for (int nt = 0; nt < 8; ++nt) acc[nt] = zero;

    for (int p = 0; p < 4; ++p) {
        // Cooperative stage of W_proj[:, p*64 .. p*64+63] into LDS (128x64, stride 68)
#pragma unroll
        for (int i = 0; i < 8; ++i) {
            const int idx = tid + i * 256;   // float4 index, 0..2047
            const int n   = idx >> 4;
            const int c4  = idx & 15;
            const float4 w4 = *reinterpret_cast<const float4*>(
                Wp + (size_t)n * CCOARSE + p * 64 + c4 * 4);
            *reinterpret_cast<float4*>(&wlds[n * LDS_STRIDE + c4 * 4]) = w4;
        }
        __syncthreads();

#pragma unroll
        for (int kc = 0; kc < 16; ++kc) {
            const int k0 = p * 64 + kc * 4 + 2 * hi;
            const v2f a = *reinterpret_cast<const v2f*>(arow + k0);

            // Issue ALL B-fragment LDS loads first, then run the 8 WMMAs
            // back-to-back so a single dscnt wait covers the whole batch.
            v2f bfrag[8];
#pragma unroll
            for (int nt = 0; nt < 8; ++nt) {
                bfrag[nt] = *reinterpret_cast<const v2f*>(
                    &wlds[(nt * 16 + l16) * LDS_STRIDE + kc * 4 + 2 * hi]);
            }
#pragma unroll
            for (int nt = 0; nt < 8; ++nt) {
                acc[nt] = __builtin_amdgcn_wmma_f32_16x16x4_f32(
                    false, a, false, bfrag[nt], (short)0, acc[nt], false, false);
            }
        }
        __syncthreads();
    }

    // Epilogue: bias + store (row = rbase + 8*hi + v, col = nt*16 + l16)
#pragma unroll
    for (int nt = 0; nt < 8; ++nt) {
        const int col  = nt * 16 + l16;
        const float bb = bp[col];
        float* outp = cproj + (size_t)(rbase + 8 * hi) * CFINE + col;
#pragma unroll
        for (int v = 0; v < 8; ++v) {
            outp[(size_t)v * CFINE] = acc[nt][v] + bb;
        }
    }
}

// ---------------------------------------------------------------------------
// Kernel B: merged[102400,128] =
//   concat(c_proj[m], unfold_gather[m,w]) [*,256] @ W_merge^T + b_merge
// row r = m*25 + w ; m in [0,4096) ; w = i*5+j
// ---------------------------------------------------------------------------
__global__ void __launch_bounds__(256)
merge_gemm_kernel(const float* __restrict__ feat_f0,
                  const float* __restrict__ feat_f1,
                  const int*   __restrict__ b_ids,
                  const int*   __restrict__ l_ids,
                  const int*   __restrict__ s_ids,
                  const float* __restrict__ cproj,
                  const float* __restrict__ Wm,
                  const float* __restrict__ bm,
                  float*       __restrict__ out)
{
    __shared__ float wlds[128 * LDS_STRIDE];

    const int tid   = threadIdx.x;
    const int wv    = tid >> 5;
    const int lane  = tid & 31;
    const int l16   = lane & 15;
    const int hi    = lane >> 4;
    const int rbase = blockIdx.x * 128 + wv * 16;
    const int row   = rbase + l16;          // [0, 102400)

    // Decode (m, window position) for this lane's A row
    const int m  = row / WW;
    const int w  = row - m * WW;
    const int ii = w / WIN;
    const int jj = w - ii * WIN;

    const float* cprow = cproj + (size_t)m * CFINE;

    int b, pos;
    const float* fsrc;
    if (m < NM) { b = b_ids[m];      pos = l_ids[m];      fsrc = feat_f0; }
    else        { const int mm = m - NM;
                  b = b_ids[mm];     pos = s_ids[mm];     fsrc = feat_f1; }

    const int y  = pos / WOUT;
    const int x  = pos - y * WOUT;
    const int h  = y * STRIDE_U + ii - PAD_U;
    const int wc = x * STRIDE_U + jj - PAD_U;
    const float maskf = (h >= 0 && h < HF && wc >= 0 && wc < WF) ? 1.0f : 0.0f;
    const int hcl = h  < 0 ? 0 : (h  > HF - 1 ? HF - 1 : h);
    const int wcl = wc < 0 ? 0 : (wc > WF - 1 ? WF - 1 : wc);
    const float* fbase = fsrc + (size_t)b * CFINE * (HF * WF)
                              + (size_t)hcl * WF + (size_t)wcl;

    v8f acc[8];
    const v8f zero = {};
#pragma unroll
    for (int nt = 0; nt < 8; ++nt) acc[nt] = zero;

    for (int p = 0; p < 4; ++p) {
        // Stage W_merge[:, p*64 .. p*64+63] into LDS
#pragma unroll
        for (int i = 0; i < 8; ++i) {
            const int idx = tid + i * 256;
            const int n   = idx >> 4;
            const int c4  = idx & 15;
            const float4 w4 = *reinterpret_cast<const float4*>(
                Wm + (size_t)n * CCOARSE + p * 64 + c4 * 4);
            *reinterpret_cast<float4*>(&wlds[n * LDS_STRIDE + c4 * 4]) = w4;
        }
        __syncthreads();

        const bool coarse_phase = (p < 2);   // k in [0,128) -> c_proj half
#pragma unroll
        for (int kc = 0; kc < 16; ++kc) {
            const int k0 = p * 64 + kc * 4 + 2 * hi;
            v2f a;
            if (coarse_phase) {
                a = *reinterpret_cast<const v2f*>(cprow + k0);
            } else {
                const int ch = k0 - CFINE;   // fine-feature channel (stride HF*WF)
                a.x = fbase[(size_t)ch       * (HF * WF)] * maskf;
                a.y = fbase[(size_t)(ch + 1) * (HF * WF)] * maskf;
            }

            v2f bfrag[8];
#pragma unroll
            for (int nt = 0; nt < 8; ++nt) {
                bfrag[nt] = *reinterpret_cast<const v2f*>(
                    &wlds[(nt * 16 + l16) * LDS_STRIDE + kc * 4 + 2 * hi]);
            }
#pragma unroll
            for (int nt = 0; nt < 8; ++nt) {
                acc[nt] = __builtin_amdgcn_wmma_f32_16x16x4_f32(
                    false, a, false, bfrag[nt], (short)0, acc[nt], false, false);
            }
        }
        __syncthreads();
    }

    // Epilogue: bias + non-temporal streaming stores
#pragma unroll
    for (int nt = 0; nt < 8; ++nt) {
        const int col  = nt * 16 + l16;
        const float bb = bm[col];
        float* outp = out + (size_t)(rbase + 8 * hi) * CFINE + col;
#pragma unroll
        for (int v = 0; v < 8; ++v) {
            __builtin_nontemporal_store(acc[nt][v] + bb, outp + (size_t)v * CFINE);
        }
    }
}

// ---------------------------------------------------------------------------
extern "C" void kernel_launch(void* const* d_in, const int* in_sizes, int n_in,
                              void* d_out, int out_size, void* d_ws, size_t ws_size,
                              hipStream_t stream)
{
    const float* feat_f0 = (const float*)d_in[0];
    const float* feat_f1 = (const float*)d_in[1];
    const float* feat_c0 = (const float*)d_in[2];
    const float* feat_c1 = (const float*)d_in[3];
    const int*   b_ids   = (const int*)d_in[4];
    const int*   l_ids   = (const int*)d_in[5];
    const int*   s_ids   = (const int*)d_in[6];
    const float* W_proj  = (const float*)d_in[7];
    const float* b_proj  = (const float*)d_in[8];
    const float* W_merge = (const float*)d_in[9];
    const float* b_merge = (const float*)d_in[10];

    float* out   = (float*)d_out;
    float* cproj = (float*)d_ws;     // 4096 x 128 fp32 = 2 MB scratch

    const int L = 4800, S = 4800;

    // Kernel A: 4096 rows / 128 rows-per-block = 32 blocks
    proj_gemm_kernel<<<32, 256, 0, stream>>>(
        feat_c0, feat_c1, b_ids, l_ids, s_ids, W_proj, b_proj, cproj, L, S);

    // Kernel B: 102400 rows / 128 rows-per-block = 800 blocks
    merge_gemm_kernel<<<800, 256, 0, stream>>>(
        feat_f0, feat_f1, b_ids, l_ids, s_ids, cproj, W_merge, b_merge, out);
}